// WTDRLoss_17729624998291
// MI455X (gfx1250) — compile-verified
//
#include <hip/hip_runtime.h>
#include <hip/hip_bf16.h>
#include <math.h>

// CDNA5 / gfx1250 fused weighted-triplet-loss.
// n = 4096, d = 128 (reference shapes). Fully fused: GEMM tiles via
// V_WMMA_F32_16X16X4_F32, distance + masked max/min mining in the epilogue,
// no 4096x4096 distance matrices ever touch memory.

typedef __attribute__((ext_vector_type(2))) float v2f;
typedef __attribute__((ext_vector_type(8))) float v8f;

#define EPSD 1e-12f
#define MARGIN 0.3f

static __device__ __forceinline__ v8f wmma_f32_16x16x4(v8f acc, v2f a, v2f b) {
    // 8 args: (neg_a, A, neg_b, B, c_mod, C, reuse_a, reuse_b)
    return __builtin_amdgcn_wmma_f32_16x16x4_f32(
        false, a, false, b, (short)0, acc, false, false);
}

// ---------------------------------------------------------------- init ----
__global__ __launch_bounds__(256) void init_stats_kernel(
    unsigned int* __restrict__ maxs, unsigned int* __restrict__ mins, int total)
{
    int i = blockIdx.x * blockDim.x + threadIdx.x;
    if (i < total) {
        maxs[i] = 0u;           // 0.0f  (all dists >= 1e-6)
        mins[i] = 0x7F800000u;  // +inf
    }
}

// --------------------------------------------------------------- norms ----
// One wave per row: |x_row|^2 via float4 loads + shfl_xor reduction.
__global__ __launch_bounds__(256) void norms_kernel(
    const float* __restrict__ inputs, float* __restrict__ norms,
    int rows, int d)
{
    int gwave = (blockIdx.x * blockDim.x + threadIdx.x) >> 5;
    int lane  = threadIdx.x & 31;
    if (gwave >= rows) return;
    const float* p = inputs + (size_t)gwave * d;
    float s = 0.0f;
    for (int k = lane * 4; k < d; k += 128) {
        float4 v = *(const float4*)(p + k);
        s += v.x * v.x + v.y * v.y + v.z * v.z + v.w * v.w;
    }
    #pragma unroll
    for (int off = 16; off >= 1; off >>= 1)
        s += __shfl_xor(s, off, 32);
    if (lane == 0) norms[gwave] = s;
}

// ----------------------------------------------------------- main pass ----
// grid = (n/16 row tiles, 3 pairs), block = 128 (4 waves).
// Each wave owns the full 16-row tile and strides over column tiles.
// dist2 = aa + bb - 2*dot; the -2 is folded into the A fragments.
__global__ __launch_bounds__(128) void pair_stats_kernel(
    const float* __restrict__ inputs, const int* __restrict__ targets,
    const float* __restrict__ norms,
    unsigned int* __restrict__ maxs, unsigned int* __restrict__ mins,
    int n, int d)
{
    const int lane = threadIdx.x & 31;
    const int wave = threadIdx.x >> 5;
    const int pair = blockIdx.y;

    // pair -> (row block, col block) base rows inside `inputs`
    int arow0, brow0;
    if (pair == 0)      { arow0 = 0;     brow0 = n;     }  // dist(x1,x2)
    else if (pair == 1) { arow0 = n;     brow0 = 2 * n; }  // dist(x2,x3)
    else                { arow0 = 0;     brow0 = 2 * n; }  // dist(x1,x3)

    const int row0 = blockIdx.x * 16;
    const int r    = lane & 15;   // fragment row
    const int kh   = lane >> 4;   // K-half selector

    // ---- A fragments (16 x K), scaled by -2, held in VGPRs for all tiles
    // Lane L holds A[r][k0 + 2*kh .. +1] per the 16x4 fp32 layout.
    v2f afrag[32];  // d = 128 -> 32 K-steps of 4
    {
        const float* aptr = inputs + (size_t)(arow0 + row0 + r) * d + 2 * kh;
        #pragma unroll
        for (int kt = 0; kt < 32; ++kt) {
            v2f t = *(const v2f*)(aptr + kt * 4);
            afrag[kt] = t * (-2.0f);
        }
    }

    // Per-lane row metadata for this lane's 8 C rows (M = v + 8*kh).
    int   trow[8];
    float aa[8];
    #pragma unroll
    for (int v = 0; v < 8; ++v) {
        int gr  = row0 + v + 8 * kh;
        trow[v] = targets[gr];
        aa[v]   = norms[arow0 + gr];
    }

    float runmax[8], runmin[8];
    #pragma unroll
    for (int v = 0; v < 8; ++v) { runmax[v] = 0.0f; runmin[v] = __int_as_float(0x7F800000); }

    const int ntiles = n >> 4;
    for (int ct = wave; ct < ntiles; ct += 4) {
        const int col0 = ct * 16;
        const int cn   = col0 + r;              // this lane's C column
        const int tcol = targets[cn];
        const float bb = norms[brow0 + cn];

        // B fragment loader: same pair layout as A (4x16, row K striped).
        const float* bptr = inputs + (size_t)(brow0 + col0 + r) * d + 2 * kh;

        v8f acc0 = {};
        v8f acc1 = {};
        #pragma unroll
        for (int kt = 0; kt < 32; kt += 2) {
            v2f b0 = *(const v2f*)(bptr + kt * 4);
            v2f b1 = *(const v2f*)(bptr + kt * 4 + 4);
            acc0 = wmma_f32_16x16x4(acc0, afrag[kt],     b0);
            acc1 = wmma_f32_16x16x4(acc1, afrag[kt + 1], b1);
        }

        // Epilogue: dist tile -> per-lane running masked max / min.
        #pragma unroll
        for (int v = 0; v < 8; ++v) {
            float m2dot = acc0[v] + acc1[v];          // = -2 * (a . b)
            float d2    = aa[v] + bb + m2dot;
            float dist  = sqrtf(fmaxf(d2, EPSD));
            if (trow[v] == tcol) runmax[v] = fmaxf(runmax[v], dist);
            else                 runmin[v] = fminf(runmin[v], dist);
        }
    }

    // Reduce across the 16 lanes of each half (xor<=8 stays in-half),
    // then lanes 0 / 16 publish rows v / v+8 with bit-pattern atomics.
    #pragma unroll
    for (int v = 0; v < 8; ++v) {
        float mx = runmax[v];
        float mn = runmin[v];
        #pragma unroll
        for (int off = 8; off >= 1; off >>= 1) {
            mx = fmaxf(mx, __shfl_xor(mx, off, 32));
            mn = fminf(mn, __shfl_xor(mn, off, 32));
        }
        if (r == 0) {
            int gr = row0 + v + 8 * kh;
            atomicMax(&maxs[pair * n + gr], __float_as_uint(mx));
            atomicMin(&mins[pair * n + gr], __float_as_uint(mn));
        }
    }
}

// ------------------------------------------------------------ finalize ----
__global__ __launch_bounds__(256) void finalize_kernel(
    const unsigned int* __restrict__ maxs, const unsigned int* __restrict__ mins,
    float* __restrict__ out, int n)
{
    __shared__ float s_sw[256];
    __shared__ float s_swl[256];
    __shared__ int   s_c[256];
    const int tid = threadIdx.x;

    float sw = 0.0f, swl = 0.0f;
    int corr = 0;
    for (int i = tid; i < n; i += 256) {
        float ap[3], an[3];
        ap[0] = __uint_as_float(maxs[0 * n + i]); an[0] = __uint_as_float(mins[2 * n + i]);
        ap[1] = __uint_as_float(maxs[1 * n + i]); an[1] = __uint_as_float(mins[0 * n + i]);
        ap[2] = __uint_as_float(maxs[2 * n + i]); an[2] = __uint_as_float(mins[1 * n + i]);
        #pragma unroll
        for (int k = 0; k < 3; ++k) {
            float loss = fmaxf(ap[k] - an[k] + MARGIN, 0.0f);
            float w = expf(loss);
            sw  += w;
            swl += w * loss;
            corr += (an[k] >= ap[k]) ? 1 : 0;
        }
    }
    s_sw[tid] = sw; s_swl[tid] = swl; s_c[tid] = corr;
    __syncthreads();
    for (int s = 128; s >= 1; s >>= 1) {
        if (tid < s) {
            s_sw[tid]  += s_sw[tid + s];
            s_swl[tid] += s_swl[tid + s];
            s_c[tid]   += s_c[tid + s];
        }
        __syncthreads();
    }
    if (tid == 0) {
        float denom = s_sw[0];
        out[0] = 3.0f * (s_swl[0] / denom);
        out[1] = (float)s_c[0];
    }
}

// ---------------------------------------------------------------- host ----
extern "C" void kernel_launch(void* const* d_in, const int* in_sizes, int n_in,
                              void* d_out, int out_size, void* d_ws, size_t ws_size,
                              hipStream_t stream) {
    const float* inputs  = (const float*)d_in[0];
    const int*   targets = (const int*)d_in[1];
    const int n = in_sizes[1];                  // 4096
    const int d = in_sizes[0] / (3 * n);        // 128
    float* out = (float*)d_out;

    // workspace: norms[3n] f32 | maxs[3n] u32 | mins[3n] u32
    float*        norms = (float*)d_ws;
    unsigned int* maxs  = (unsigned int*)(norms + 3 * n);
    unsigned int* mins  = maxs + 3 * n;

    const int total = 3 * n;
    init_stats_kernel<<<(total + 255) / 256, 256, 0, stream>>>(maxs, mins, total);
    // one wave per row -> 8 rows per 256-thread block
    norms_kernel<<<(total + 7) / 8, 256, 0, stream>>>(inputs, norms, total, d);
    dim3 grid(n / 16, 3);
    pair_stats_kernel<<<grid, 128, 0, stream>>>(inputs, targets, norms,
                                                maxs, mins, n, d);
    finalize_kernel<<<1, 256, 0, stream>>>(maxs, mins, out, n);
}